// GraphConvolutionLayer_11141145166348
// MI455X (gfx1250) — compile-verified
//
#include <hip/hip_runtime.h>

typedef float v2f __attribute__((ext_vector_type(2)));
typedef float v8f __attribute__((ext_vector_type(8)));

#define BDIM 4
#define NN   8192
#define KNEI 20
#define INS  256
#define HD   256
#define ED   5
#define AGG_STRIDE 260   // 256 + 4 pad -> 16 rows map to distinct LDS banks

// ---------------------------------------------------------------------------
// Kernel A: W_comb[e] = W @ W_edge[e]^T, stored PRE-SWIZZLED into the exact
// WMMA B-fragment layout kernel B consumes:
//   frag(e, ntile, f) covers B[k = 4f .. 4f+3][n = 16*ntile .. 16*ntile+15]
//   value B[k][n] lives at:
//     Wc[((e*16 + ntile)*64 + f)*64 + ls*2 + v]
//     ls = ((k>>1)&1)*16 + (n&15),  v = k&1          (lane ls, dword v)
// so kernel B loads one fragment with a single coalesced global_load_b64.
// One wave per 16x16 output tile; 5*16*16 = 1280 wave-jobs.
// ---------------------------------------------------------------------------
__global__ void __launch_bounds__(256)
wcomb_kernel(const float* __restrict__ W,
             const float* __restrict__ WE,
             float* __restrict__ Wc) {
  const int wid  = threadIdx.x >> 5;
  const int lane = threadIdx.x & 31;
  const int job  = blockIdx.x * 8 + wid;        // 0..1279
  const int e    = job >> 8;
  const int rem  = job & 255;
  const int i0   = (rem >> 4) << 4;             // K-block of W_comb (i index)
  const int o0   = (rem & 15) << 4;             // N-block of W_comb (o index)

  const int m  = lane & 15;                     // A: M row / B: N col
  const int kh = (lane >> 4) << 1;              // K sub-offset per lane half

  // A[m][k] = W[i0+m][k]           (contiguous pair in k -> b64 load)
  const float* Arow = W  + (size_t)(i0 + m) * HD;
  // B[k][n] = WE[e][o0+n][k]       (contiguous pair in k -> b64 load)
  const float* Brow = WE + ((size_t)e * HD + (o0 + m)) * HD;

  v8f c = {};
#pragma unroll 8
  for (int k0 = 0; k0 < HD; k0 += 4) {
    v2f a = *(const v2f*)(Arow + k0 + kh);
    v2f b = *(const v2f*)(Brow + k0 + kh);
    c = __builtin_amdgcn_wmma_f32_16x16x4_f32(false, a, false, b,
                                              (short)0, c, false, false);
  }

  // scatter D tile into B-fragment layout (i plays the role of k in kernel B)
  const int col   = lane & 15;
  const int rbase = (lane >> 4) << 3;           // lanes 16-31 hold rows M=8..15
  const size_t tilebase = (size_t)(e * 16 + (o0 >> 4)) * 64;   // frag group
#pragma unroll
  for (int r = 0; r < 8; ++r) {
    const int i  = i0 + rbase + r;              // k index in kernel B's GEMM
    const int f  = i >> 2;                      // fragment id (k group of 4)
    const int ls = (((i >> 1) & 1) << 4) + col; // target lane
    const int v  = i & 1;                       // target dword within lane
    Wc[(tilebase + f) * 64 + ls * 2 + v] = c[r];
  }
}

// ---------------------------------------------------------------------------
// Kernel B: fused neighbor gather/aggregate + per-edge-type GEMM + epilogue.
// One 256-thread block (8 waves) per 16-node tile.
//   Phase 1: wave w aggregates nodes 2w, 2w+1: per-edge-type register sums of
//            neighbor X rows (neighbor idx is wave-uniform -> scalar-base
//            loads), then writes agg[e][m][0..255] into padded LDS.
//   Phase 2: wave w computes output N-tiles 2w, 2w+1 together:
//            one A-fragment ds_load_b64 feeds TWO wmma (two accumulators),
//            B-fragments are single coalesced global_load_b64 from the
//            pre-swizzled W_comb. Then /deg + bias + relu.
// ---------------------------------------------------------------------------
__global__ void __launch_bounds__(256)
gcn_gather_gemm(const float* __restrict__ X,
                const int*   __restrict__ nnei,
                const int*   __restrict__ nedge,
                const float* __restrict__ Wc,
                const float* __restrict__ bias,
                float*       __restrict__ Y) {
  extern __shared__ float lds[];
  float* agg  = lds;                            // [ED][16][AGG_STRIDE]
  float* degs = lds + ED * 16 * AGG_STRIDE;     // [16]

  const int lane      = threadIdx.x & 31;
  const int wid       = threadIdx.x >> 5;
  const int bid       = blockIdx.x;
  const int b         = bid >> 9;               // 512 node-tiles per batch
  const int node_base = (bid & 511) << 4;

  // ---------------- Phase 1: gather + aggregate ----------------
#pragma unroll
  for (int mi = 0; mi < 2; ++mi) {
    const int m    = (wid << 1) + mi;
    const int node = node_base + m;
    const size_t ibase = ((size_t)b * NN + node) * KNEI;

    int nk = -1, ek = -1;
    if (lane < KNEI) {
      nk = nnei[ibase + lane];
      ek = nedge[ibase + lane];
    }
    const unsigned long long bal = __ballot(lane < KNEI && ek >= 0);
    int deg = (int)__popcll(bal);
    if (deg < 1) deg = 1;

    float4 acc[ED][2];
#pragma unroll
    for (int e = 0; e < ED; ++e) {
      acc[e][0] = make_float4(0.f, 0.f, 0.f, 0.f);
      acc[e][1] = make_float4(0.f, 0.f, 0.f, 0.f);
    }

    for (int k = 0; k < KNEI; ++k) {
      // broadcast slot k -> wave-uniform; force into SGPRs
      const int nei = __builtin_amdgcn_readfirstlane(__shfl(nk, k, 32));
      if (nei < 0) continue;
      const int et  = __builtin_amdgcn_readfirstlane(__shfl(ek, k, 32));
      const float4* row = (const float4*)(X + ((size_t)b * NN + nei) * INS);
      const float4 x0 = row[lane];              // cols   4*lane .. 4*lane+3
      const float4 x1 = row[lane + 32];         // cols 128+4*lane ..
#pragma unroll
      for (int e = 0; e < ED; ++e) {
        if (et == e) {                          // uniform scalar branch
          acc[e][0].x += x0.x; acc[e][0].y += x0.y;
          acc[e][0].z += x0.z; acc[e][0].w += x0.w;
          acc[e][1].x += x1.x; acc[e][1].y += x1.y;
          acc[e][1].z += x1.z; acc[e][1].w += x1.w;
        }
      }
    }

#pragma unroll
    for (int e = 0; e < ED; ++e) {
      float* dst = agg + ((size_t)(e * 16 + m)) * AGG_STRIDE;
      *(float4*)(dst + lane * 4)       = acc[e][0];
      *(float4*)(dst + 128 + lane * 4) = acc[e][1];
    }
    if (lane == 0) degs[m] = (float)deg;
  }
  __syncthreads();

  // ---------------- Phase 2: WMMA GEMM (2 tiles/wave) ----------------
  const int am = lane & 15;                     // A row
  const int kh = (lane >> 4) << 1;              // K sub-offset per lane half
  const int t0 = wid << 1;                      // first N-tile of this wave

  v8f c0 = {};
  v8f c1 = {};
  for (int e = 0; e < ED; ++e) {
    const float* Ab = agg + (size_t)(e * 16 + am) * AGG_STRIDE;
    const float* B0 = Wc + (size_t)(e * 16 + t0) * 64 * 64 + lane * 2;
    const float* B1 = B0 + (size_t)64 * 64;     // next N-tile's fragments
#pragma unroll 8
    for (int kg = 0; kg < 64; ++kg) {
      v2f a  = *(const v2f*)(Ab + (kg << 2) + kh);    // ds_load_b64
      v2f b0 = *(const v2f*)(B0 + (kg << 6));         // coalesced b64
      v2f b1 = *(const v2f*)(B1 + (kg << 6));         // coalesced b64
      c0 = __builtin_amdgcn_wmma_f32_16x16x4_f32(false, a, false, b0,
                                                 (short)0, c0, false, false);
      c1 = __builtin_amdgcn_wmma_f32_16x16x4_f32(false, a, false, b1,
                                                 (short)0, c1, false, false);
    }
  }

  // ---------------- Epilogue: /deg + bias + relu ----------------
  const int col   = lane & 15;
  const int rbase = (lane >> 4) << 3;
#pragma unroll
  for (int ti = 0; ti < 2; ++ti) {
    const v8f c  = ti ? c1 : c0;
    const int n0 = (t0 + ti) << 4;
    const float bc = bias[n0 + col];
#pragma unroll
    for (int r = 0; r < 8; ++r) {
      const int row = rbase + r;
      float v = c[r] / degs[row] + bc;
      v = v > 0.f ? v : 0.f;
      Y[((size_t)b * NN + node_base + row) * HD + n0 + col] = v;
    }
  }
}

// ---------------------------------------------------------------------------
extern "C" void kernel_launch(void* const* d_in, const int* in_sizes, int n_in,
                              void* d_out, int out_size, void* d_ws, size_t ws_size,
                              hipStream_t stream) {
  const float* X    = (const float*)d_in[0];   // [B,N,IN]
  const float* W    = (const float*)d_in[1];   // [IN,H]
  const float* WE   = (const float*)d_in[2];   // [E,H,H]
  const float* bias = (const float*)d_in[3];   // [H]
  const int*   nnei = (const int*)d_in[4];     // [B,N,K]
  const int*   nedg = (const int*)d_in[5];     // [B,N,K]
  float*       Y    = (float*)d_out;           // [B,N,H]
  float*       Wc   = (float*)d_ws;            // swizzled W_comb, 1.25 MB

  (void)in_sizes; (void)n_in; (void)out_size; (void)ws_size;

  // Kernel A: 1280 wave-jobs, 8 waves per block -> 160 blocks
  wcomb_kernel<<<160, 256, 0, stream>>>(W, WE, Wc);

  // Kernel B: one block per 16-node tile
  const size_t ldsB = (size_t)(ED * 16 * AGG_STRIDE + 16) * sizeof(float);
  gcn_gather_gemm<<<BDIM * (NN / 16), 256, ldsB, stream>>>(X, nnei, nedg, Wc, bias, Y);
}